// MaskedChamferL1Loss_66159676228324
// MI455X (gfx1250) — compile-verified
//
#include <hip/hip_runtime.h>

typedef __attribute__((ext_vector_type(2))) float v2f;
typedef __attribute__((ext_vector_type(8))) float v8f;

#define B_  4
#define N_  8192
#define BIG_ 1e9f

// ---------------------------------------------------------------------------
// Phase 1: build packed per-point data (two forms per point set).
//   A-form: ( x,  y,  z,  ||.||^2 + pen)   -> WMMA A fragment (row side)
//   B-form: (-2x, -2y, -2z, ||.||^2 + pen) -> WMMA B fragment (col side)
// pen = (1 - mask) * 1e9 folded into the norm term:
//   max(d2,0)+pen == max(d2+pen,0) for all entries that can ever win the min,
//   and penalized rows are zeroed by the mask weight in the finalize.
// ---------------------------------------------------------------------------
__global__ __launch_bounds__(256)
void prep_kernel(const float* __restrict__ pred, const float* __restrict__ target,
                 const int* __restrict__ mask, const float* __restrict__ points,
                 float4* __restrict__ cleanA, float4* __restrict__ cleanB,
                 float4* __restrict__ predA,  float4* __restrict__ predB) {
  int i = blockIdx.x * blockDim.x + threadIdx.x;
  if (i >= B_ * N_) return;
  int base = i * 3;
  float cx = points[base + 0] + target[base + 0];
  float cy = points[base + 1] + target[base + 1];
  float cz = points[base + 2] + target[base + 2];
  float px = points[base + 0] + pred[base + 0];
  float py = points[base + 1] + pred[base + 1];
  float pz = points[base + 2] + pred[base + 2];
  float pen = (1.0f - (float)mask[i]) * BIG_;
  float cw = cx * cx + cy * cy + cz * cz + pen;
  float pw = px * px + py * py + pz * pz + pen;
  cleanA[i] = make_float4(cx, cy, cz, cw);
  cleanB[i] = make_float4(-2.0f * cx, -2.0f * cy, -2.0f * cz, cw);
  predA[i]  = make_float4(px, py, pz, pw);
  predB[i]  = make_float4(-2.0f * px, -2.0f * py, -2.0f * pz, pw);
}

// ---------------------------------------------------------------------------
// Phase 2: directed Chamfer min via V_WMMA_F32_16X16X4_F32.
// One wave32 per 16-row strip of A. For every 16-col tile of B:
//   A frag (16x4): row m -> (a_x, a_y, a_z, wa)        wa = ||a||^2 + pen_a
//   B frag (4x16): col n -> (-2b_x, -2b_y, -2b_z, 1)
//   WMMA -> S[m][n] = -2 a.b + wa ; d = max(S + wb, 0)
// Running min per (row,lane-column), then xor-shuffle row reduction.
// minOut[b*N + row] = min_n d  — each row owned by exactly one wave.
// ---------------------------------------------------------------------------
__global__ __launch_bounds__(32)
void chamfer_min_kernel(const float4* __restrict__ Aq, const float4* __restrict__ Bq,
                        float* __restrict__ minOut) {
  const int strips = N_ / 16;
  const int b    = blockIdx.x / strips;
  const int n0   = (blockIdx.x % strips) * 16;
  const int lane = threadIdx.x;
  const int half = lane >> 4;   // 0: K=0,1 / rows j  | 1: K=2,3 / rows j+8
  const int mrow = lane & 15;   // A-row index within strip; also tile column

  // A fragment: 16x4 f32 layout — lanes 0-15: K=0,1 ; lanes 16-31: K=2,3
  float4 a4 = Aq[b * N_ + n0 + mrow];
  v2f afrag;
  afrag[0] = half ? a4.z : a4.x;
  afrag[1] = half ? a4.w : a4.y;

  v8f rmin;
#pragma unroll
  for (int j = 0; j < 8; ++j) rmin[j] = 3.0e38f;

  const float4* Bb = Bq + b * N_;

#pragma unroll 4
  for (int j0 = 0; j0 < N_; j0 += 16) {
    float4 b4 = Bb[j0 + mrow];   // this lane's column (prescaled by -2, w in .w)

    // B fragment: 4x16 f32 — VGPR0: K=(half?2:0), VGPR1: K=(half?3:1)
    v2f bfrag;
    bfrag[0] = half ? b4.z : b4.x;
    bfrag[1] = half ? 1.0f : b4.y;

    v8f acc = {};
    acc = __builtin_amdgcn_wmma_f32_16x16x4_f32(
        /*neg_a=*/false, afrag, /*neg_b=*/false, bfrag,
        /*c_mod=*/(short)0, acc, /*reuse_a=*/false, /*reuse_b=*/false);

#pragma unroll
    for (int j = 0; j < 8; ++j) {
      float d = fmaxf(acc[j] + b4.w, 0.0f);  // + (||b||^2 + pen), clamp
      rmin[j] = fminf(rmin[j], d);
    }
  }

  // Row reduction: min across the 16 lane-columns of each half-wave.
#pragma unroll
  for (int j = 0; j < 8; ++j) {
#pragma unroll
    for (int off = 1; off < 16; off <<= 1)
      rmin[j] = fminf(rmin[j], __shfl_xor(rmin[j], off, 32));
  }

  if (mrow == 0) {
#pragma unroll
    for (int j = 0; j < 8; ++j)
      minOut[b * N_ + n0 + half * 8 + j] = rmin[j];
  }
}

// ---------------------------------------------------------------------------
// Phase 3: masked L1 + per-batch Chamfer normalization -> scalar loss.
// ---------------------------------------------------------------------------
__device__ __forceinline__ float block_reduce_sum(float v, float* s) {
#pragma unroll
  for (int off = 16; off > 0; off >>= 1) v += __shfl_xor(v, off, 32);
  int wid = threadIdx.x >> 5;
  if ((threadIdx.x & 31) == 0) s[wid] = v;
  __syncthreads();
  if (threadIdx.x == 0) {
    float r = s[0];
#pragma unroll
    for (int i = 1; i < 8; ++i) r += s[i];
    s[0] = r;
  }
  __syncthreads();
  float r = s[0];
  __syncthreads();
  return r;
}

__global__ __launch_bounds__(256)
void finalize_kernel(const float* __restrict__ minx, const float* __restrict__ miny,
                     const int* __restrict__ mask, const float* __restrict__ pred,
                     const float* __restrict__ target, float* __restrict__ out) {
  __shared__ float s[8];
  __shared__ float chamAcc;
  const int tid = threadIdx.x;
  if (tid == 0) chamAcc = 0.0f;
  __syncthreads();

  float l1s = 0.0f, ms = 0.0f;
  for (int b = 0; b < B_; ++b) {
    float sx = 0.0f, sy = 0.0f, c = 0.0f;
    for (int n = tid; n < N_; n += 256) {
      int i = b * N_ + n;
      float m = (float)mask[i];
      sx += minx[i] * m;
      sy += miny[i] * m;
      c  += m;
      int base = i * 3;
      float d0 = fabsf(pred[base + 0] - target[base + 0]);
      float d1 = fabsf(pred[base + 1] - target[base + 1]);
      float d2 = fabsf(pred[base + 2] - target[base + 2]);
      l1s += m * (d0 + d1 + d2) * (1.0f / 3.0f);
      ms  += m;
    }
    sx = block_reduce_sum(sx, s);
    sy = block_reduce_sum(sy, s);
    c  = block_reduce_sum(c, s);
    if (tid == 0) chamAcc += (sx + sy) / fmaxf(c, 1.0f);
    __syncthreads();
  }
  l1s = block_reduce_sum(l1s, s);
  ms  = block_reduce_sum(ms, s);
  if (tid == 0) out[0] = 0.5f * (l1s / ms + chamAcc / (float)B_);
}

// ---------------------------------------------------------------------------
// Workspace layout (floats):
//   cleanA[4BN] | cleanB[4BN] | predA[4BN] | predB[4BN] | minx[BN] | miny[BN]
// = 18 * B*N * 4 B ≈ 2.36 MB
// ---------------------------------------------------------------------------
extern "C" void kernel_launch(void* const* d_in, const int* in_sizes, int n_in,
                              void* d_out, int out_size, void* d_ws, size_t ws_size,
                              hipStream_t stream) {
  const float* pred   = (const float*)d_in[0];
  const float* target = (const float*)d_in[1];
  const int*   mask   = (const int*)d_in[2];
  const float* points = (const float*)d_in[3];
  float* out = (float*)d_out;

  float*  ws     = (float*)d_ws;
  float4* cleanA = (float4*)(ws + 0  * B_ * N_);
  float4* cleanB = (float4*)(ws + 4  * B_ * N_);
  float4* predA  = (float4*)(ws + 8  * B_ * N_);
  float4* predB  = (float4*)(ws + 12 * B_ * N_);
  float*  minx   = ws + 16 * B_ * N_;
  float*  miny   = ws + 17 * B_ * N_;

  prep_kernel<<<(B_ * N_ + 255) / 256, 256, 0, stream>>>(
      pred, target, mask, points, cleanA, cleanB, predA, predB);

  // min over pred for each clean row, then the symmetric direction.
  chamfer_min_kernel<<<B_ * (N_ / 16), 32, 0, stream>>>(cleanA, predB, minx);
  chamfer_min_kernel<<<B_ * (N_ / 16), 32, 0, stream>>>(predA, cleanB, miny);

  finalize_kernel<<<1, 256, 0, stream>>>(minx, miny, mask, pred, target, out);
}